// LocallyConnected2d_87016037417130
// MI455X (gfx1250) — compile-verified
//
#include <hip/hip_runtime.h>

typedef float wf2 __attribute__((ext_vector_type(2)));
typedef float wf8 __attribute__((ext_vector_type(8)));

#define B_   16
#define CI   64
#define CO   64
#define H_   64
#define W_   64
#define KK   9          // 3x3 taps
#define WB   2          // w positions per workgroup
#define CC   4          // c channels per K-chunk (one K=4 WMMA step per tap)
#define NCH  (CI / CC)  // 16 chunks
#define NT   256        // 8 waves

#define WSTRIDE_O (CI * H_ * W_ * KK)   // 2359296
#define WSTRIDE_C (H_ * W_ * KK)        // 36864
#define WSTRIDE_H (W_ * KK)             // 576
#define XSTRIDE_B (CI * H_ * W_)        // 262144
#define XSTRIDE_C (H_ * W_)             // 4096

#define NWREG 18   // (CO*CC*WB*KK)/NT
#define NXREG 3    // (3*CC*(WB+2)*B_)/NT

// wlds physical layout: [inner=(w*9+k)][o rotated by inner][c]  (c fastest -> b64 operand loads)
// xlds physical layout: [j=(r*4+wc)][b rotated by j][c]
__global__ __launch_bounds__(NT) void lc2d_wmma(const float* __restrict__ x,
                                                const float* __restrict__ wgt,
                                                float* __restrict__ out) {
    __shared__ float wlds[WB * KK * CO * CC];        // 4608 f = 18 KB
    __shared__ float xlds[3 * (WB + 2) * B_ * CC];   //  768 f =  3 KB

    const int tid  = threadIdx.x;
    const int h    = blockIdx.x >> 5;          // / (W_/WB)
    const int w0   = (blockIdx.x & 31) * WB;

    const int lane = tid & 31;
    const int wave = tid >> 5;
    const int w_l  = wave >> 2;                // 0..1
    const int o0   = (wave & 3) * 16;          // o tile base
    const int hi   = lane >> 4;                // half-wave -> K 2/3 vs 0/1
    const int ln   = lane & 15;

    // ---- per-thread global gather offsets + swizzled LDS store offsets ----
    int wg[NWREG], ws[NWREG];
#pragma unroll
    for (int i = 0; i < NWREG; ++i) {
        int f     = tid + i * NT;
        int inner = f % (WB * KK);             // w*9+k : contiguous run in memory
        int seg   = f / (WB * KK);
        int cl    = seg & (CC - 1);
        int o     = seg >> 2;
        int wl    = inner / KK;
        int k     = inner - wl * KK;
        wg[i] = o * WSTRIDE_O + cl * WSTRIDE_C + h * WSTRIDE_H + (w0 + wl) * KK + k;
        ws[i] = inner * (CO * CC) + (((o + inner) & 63) << 2) + cl;  // rotate o by inner
    }
    int xg[NXREG], xs[NXREG];
    bool xv[NXREG];
#pragma unroll
    for (int i = 0; i < NXREG; ++i) {
        int f  = tid + i * NT;
        int wc = f & 3;
        int r  = (f >> 2) % 3;
        int cl = (f / 12) & 3;
        int b  = f / 48;
        int hh = h - 1 + r;
        int ww = w0 - 1 + wc;
        bool ok = (hh >= 0) & (hh < H_) & (ww >= 0) & (ww < W_);
        xv[i] = ok;
        xg[i] = ok ? (b * XSTRIDE_B + cl * XSTRIDE_C + hh * W_ + ww) : 0;
        int j  = r * (WB + 2) + wc;
        xs[i]  = j * (B_ * CC) + (((b + j) & 15) << 2) + cl;         // rotate b by j
    }

    // ---- chunk-invariant LDS operand addresses: compute ONCE into registers ----
    int aofs[KK], bofs[KK];
#pragma unroll
    for (int k = 0; k < KK; ++k) {
        const int kh = k / 3, kw = k % 3;
        const int j  = kh * (WB + 2) + (w_l + kw);
        aofs[k] = j * (B_ * CC) + (((ln + j) & 15) << 2) + 2 * hi;
        const int inner = w_l * KK + k;
        bofs[k] = inner * (CO * CC) + (((o0 + ln + inner) & 63) << 2) + 2 * hi;
    }

    // ---- prologue: stage chunk 0 ----
    float wtmp[NWREG], xtmp[NXREG];
#pragma unroll
    for (int i = 0; i < NWREG; ++i)
        wtmp[i] = __builtin_nontemporal_load(wgt + wg[i]);
#pragma unroll
    for (int i = 0; i < NXREG; ++i)
        xtmp[i] = xv[i] ? x[xg[i]] : 0.0f;

    wf8 acc = {};

#pragma unroll 1
    for (int ch = 0; ch < NCH; ++ch) {
        __syncthreads();
#pragma unroll
        for (int i = 0; i < NWREG; ++i) wlds[ws[i]] = wtmp[i];
#pragma unroll
        for (int i = 0; i < NXREG; ++i) xlds[xs[i]] = xtmp[i];
        __syncthreads();

        // issue NEXT chunk's global loads now (uniform chunk base), hide under WMMAs
        if (ch + 1 < NCH) {
            const float* wp = wgt + (ch + 1) * (CC * WSTRIDE_C);  // uniform
            const float* xp = x   + (ch + 1) * (CC * XSTRIDE_C);  // uniform
#pragma unroll
            for (int i = 0; i < NWREG; ++i)
                wtmp[i] = __builtin_nontemporal_load(wp + wg[i]);
#pragma unroll
            for (int i = 0; i < NXREG; ++i)
                xtmp[i] = xv[i] ? xp[xg[i]] : 0.0f;
        }

        // ---- batch all 18 ds_load_b64, then 9 dense WMMAs ----
        wf2 av[KK], bv[KK];
#pragma unroll
        for (int k = 0; k < KK; ++k) {
            av[k] = *(const wf2*)&xlds[aofs[k]];
            bv[k] = *(const wf2*)&wlds[bofs[k]];
        }
#pragma unroll
        for (int k = 0; k < KK; ++k) {
            acc = __builtin_amdgcn_wmma_f32_16x16x4_f32(
                false, av[k], false, bv[k], (short)0, acc, false, false);
        }
    }

    // ---- store C tile (scatter merges in the 192 MB L2; out fits entirely) ----
#pragma unroll
    for (int r = 0; r < 8; ++r) {
        const int bb = r + 8 * hi;
        const int oo = o0 + ln;
        out[((bb * CO + oo) * H_ + h) * W_ + (w0 + w_l)] = acc[r];
    }
}

extern "C" void kernel_launch(void* const* d_in, const int* in_sizes, int n_in,
                              void* d_out, int out_size, void* d_ws, size_t ws_size,
                              hipStream_t stream) {
    const float* x   = (const float*)d_in[0];   // (16,64,64,64) f32
    const float* wgt = (const float*)d_in[1];   // (1,64,64,64,64,9) f32
    float* out       = (float*)d_out;           // (16,64,64,64) f32
    dim3 grid(H_ * (W_ / WB));                  // 2048 workgroups
    dim3 block(NT);                             // 8 waves
    lc2d_wmma<<<grid, block, 0, stream>>>(x, wgt, out);
}